// Model_13168369730039
// MI455X (gfx1250) — compile-verified
//
#include <hip/hip_runtime.h>
#include <hip/hip_bf16.h>

// ---------------------------------------------------------------------------
// GraphRec-style model for MI455X (gfx1250, wave32, WMMA).
// All matrix math via V_WMMA_F32_16X16X4_F32 (f32-faithful to the reference).
// ---------------------------------------------------------------------------

typedef __attribute__((ext_vector_type(2))) float v2f;
typedef __attribute__((ext_vector_type(8))) float v8f;

#define HH   128
#define BB   2048
#define UU   50000
#define II   50000
#define EE1  102400
#define EE2  61440
#define EE3  102400

__device__ __forceinline__ v8f wmma_f32_k4(v2f a, v2f b, v8f c) {
  // D = A(16x4,f32) * B(4x16,f32) + C(16x16,f32)
  return __builtin_amdgcn_wmma_f32_16x16x4_f32(
      /*neg_a=*/false, a, /*neg_b=*/false, b,
      /*c_mod=*/(short)0, c, /*reuse_a=*/false, /*reuse_b=*/false);
}

__device__ __forceinline__ unsigned f2ord(float f) {
  unsigned u = __float_as_uint(f);
  return (u & 0x80000000u) ? ~u : (u | 0x80000000u);
}
__device__ __forceinline__ float ord2f(unsigned u) {
  return __uint_as_float((u & 0x80000000u) ? (u ^ 0x80000000u) : ~u);
}

// ---------------------------------------------------------------------------
// Generic gather-concat GEMM:  out[r,0:128] = act( concat(srcA[gA(r)], srcB[gB(r)]) @ W + bias )
// K = 128 (srcB unused) or 256.  M = gridDim.x*16.  W is [K,128] row-major.
// Block = 128 threads (4 waves); each wave does 2 N-tiles of 16 columns.
// ---------------------------------------------------------------------------
template <int K>
__global__ void gemm_gcat_kernel(const float* __restrict__ srcA, const int* __restrict__ idxA,
                                 const float* __restrict__ srcB, const int* __restrict__ idxB,
                                 const float* __restrict__ W, const float* __restrict__ bias,
                                 float* __restrict__ out, int relu) {
  constexpr int LD = K + 4;                 // pad -> conflict-free column reads
  __shared__ __align__(16) float ldsA[16 * LD];
  __shared__ int ldsIdx[32];

  const int tid = threadIdx.x;
  const int rowbase = blockIdx.x * 16;

  if (tid < 16) ldsIdx[tid] = idxA ? idxA[rowbase + tid] : (rowbase + tid);
  if (K == 256 && tid >= 16 && tid < 32)
    ldsIdx[tid] = idxB ? idxB[rowbase + tid - 16] : (rowbase + tid - 16);
  __syncthreads();

  for (int i = tid; i < 16 * K; i += 128) {
    const int r = i / K, c = i % K;
    float v;
    if (c < HH) v = srcA[(size_t)ldsIdx[r] * HH + c];
    else        v = srcB[(size_t)ldsIdx[16 + r] * HH + (c - HH)];
    ldsA[r * LD + c] = v;
  }
  __syncthreads();

  const int wave = tid >> 5, lane = tid & 31;
  const int ln = lane & 15, half = lane >> 4;   // A-row m == D-col n == lane%16

  for (int nt = wave; nt < 8; nt += 4) {
    const int nbase = nt * 16;
    const float bv = bias[nbase + ln];
    v8f c;
#pragma unroll
    for (int r = 0; r < 8; ++r) c[r] = bv;

#pragma unroll 4
    for (int k4 = 0; k4 < K; k4 += 4) {
      v2f a = *(const v2f*)&ldsA[ln * LD + k4 + 2 * half];
      v2f b;
      b.x = W[(size_t)(k4 + 2 * half)     * HH + nbase + ln];
      b.y = W[(size_t)(k4 + 2 * half + 1) * HH + nbase + ln];
      c = wmma_f32_k4(a, b, c);
    }
#pragma unroll
    for (int r = 0; r < 8; ++r) {
      float v = c[r];
      if (relu) v = fmaxf(v, 0.0f);
      out[(size_t)(rowbase + r + 8 * half) * HH + nbase + ln] = v;
    }
  }
}

// ---------------------------------------------------------------------------
// Fused attention score:  s[r] = relu( concat(srcA[gA(r)], srcB[gB(r)]) @ W1 + b1 ) . w2 + b2
// K=256 fixed; hidden 16x128 tile kept in LDS, never spilled to HBM.
// ---------------------------------------------------------------------------
__global__ void attn_score_kernel(const float* __restrict__ srcA, const int* __restrict__ idxA,
                                  const float* __restrict__ srcB, const int* __restrict__ idxB,
                                  const float* __restrict__ W1, const float* __restrict__ b1,
                                  const float* __restrict__ w2, const float* __restrict__ b2,
                                  float* __restrict__ outs) {
  constexpr int K = 256, LD = K + 4, LH = HH + 4;
  __shared__ __align__(16) float ldsA[16 * LD];
  __shared__ __align__(16) float hid[16 * LH];
  __shared__ int ldsIdx[32];

  const int tid = threadIdx.x;
  const int rowbase = blockIdx.x * 16;

  if (tid < 16) ldsIdx[tid] = idxA ? idxA[rowbase + tid] : (rowbase + tid);
  if (tid >= 16 && tid < 32) ldsIdx[tid] = idxB ? idxB[rowbase + tid - 16] : (rowbase + tid - 16);
  __syncthreads();

  for (int i = tid; i < 16 * K; i += 128) {
    const int r = i / K, c = i % K;
    float v;
    if (c < HH) v = srcA[(size_t)ldsIdx[r] * HH + c];
    else        v = srcB[(size_t)ldsIdx[16 + r] * HH + (c - HH)];
    ldsA[r * LD + c] = v;
  }
  __syncthreads();

  const int wave = tid >> 5, lane = tid & 31;
  const int ln = lane & 15, half = lane >> 4;

  for (int nt = wave; nt < 8; nt += 4) {
    const int nbase = nt * 16;
    const float bv = b1[nbase + ln];
    v8f c;
#pragma unroll
    for (int r = 0; r < 8; ++r) c[r] = bv;

#pragma unroll 4
    for (int k4 = 0; k4 < K; k4 += 4) {
      v2f a = *(const v2f*)&ldsA[ln * LD + k4 + 2 * half];
      v2f b;
      b.x = W1[(size_t)(k4 + 2 * half)     * HH + nbase + ln];
      b.y = W1[(size_t)(k4 + 2 * half + 1) * HH + nbase + ln];
      c = wmma_f32_k4(a, b, c);
    }
#pragma unroll
    for (int r = 0; r < 8; ++r)
      hid[(r + 8 * half) * LH + nbase + ln] = fmaxf(c[r], 0.0f);
  }
  __syncthreads();

  if (tid < 16) {
    float s = b2[0];
    for (int k = 0; k < HH; ++k) s += hid[tid * LH + k] * w2[k];
    outs[rowbase + tid] = s;
  }
}

// ---------------------------------------------------------------------------
// Segment softmax helpers (torch_scatter.softmax semantics)
// ---------------------------------------------------------------------------
__global__ void compose_idx_kernel(const int* __restrict__ outer, const int* __restrict__ inner,
                                   int* __restrict__ out, int n) {
  int e = blockIdx.x * 256 + threadIdx.x;
  if (e < n) out[e] = inner[outer[e]];
}

__global__ void seg_max_kernel(const float* __restrict__ sc, const int* __restrict__ seg,
                               unsigned* __restrict__ segmax, int E) {
  int e = blockIdx.x * 256 + threadIdx.x;
  if (e < E) atomicMax(&segmax[seg[e]], f2ord(sc[e]));
}

__global__ void seg_expsum_kernel(float* __restrict__ sc, const int* __restrict__ seg,
                                  const unsigned* __restrict__ segmax,
                                  float* __restrict__ segsum, int E) {
  int e = blockIdx.x * 256 + threadIdx.x;
  if (e < E) {
    int s = seg[e];
    float ee = expf(sc[e] - ord2f(segmax[s]));
    sc[e] = ee;
    atomicAdd(&segsum[s], ee);
  }
}

// acc[seg[e], c] += feat[g(e), c] * sc[e] / segsum[seg[e]]
__global__ void seg_scatter_kernel(const float* __restrict__ feat, const int* __restrict__ fidx,
                                   const float* __restrict__ sc, const float* __restrict__ segsum,
                                   const int* __restrict__ seg, float* __restrict__ acc, int E) {
  int g = blockIdx.x * 256 + threadIdx.x;
  if (g >= E * HH) return;
  int e = g >> 7, c = g & (HH - 1);
  int s = seg[e];
  float w = sc[e] / segsum[s];
  int row = fidx ? fidx[e] : e;
  atomicAdd(&acc[(size_t)s * HH + c], feat[(size_t)row * HH + c] * w);
}

// out[r] = A[r,:] . w + b   (one wave per row)
__global__ void rowdot_kernel(const float* __restrict__ A, const float* __restrict__ w,
                              const float* __restrict__ b, float* __restrict__ out, int M) {
  int r = blockIdx.x * blockDim.y + threadIdx.y;
  if (r >= M) return;
  int l = threadIdx.x;
  float s = 0.0f;
  for (int k = l; k < HH; k += 32) s += A[(size_t)r * HH + k] * w[k];
  for (int off = 16; off; off >>= 1) s += __shfl_down(s, off, 32);
  if (l == 0) out[r] = s + b[0];
}

// ---------------------------------------------------------------------------
// logits = h @ item_w.T   [2048,128] x [50000,128]^T, NT GEMM, f32 WMMA.
// Block = 128 threads; h-tile shared in LDS; each wave stages its 16-row
// item_w tile in LDS (coalesced float4) then runs 32 WMMA k-steps.
// ---------------------------------------------------------------------------
#define NTILES_J ((II + 15) / 16)   // 3125
__global__ void logits_kernel(const float* __restrict__ h, const float* __restrict__ item_w,
                              float* __restrict__ out) {
  constexpr int LH = HH + 4;
  __shared__ __align__(16) float ldsH[16 * LH];
  __shared__ __align__(16) float ldsB[4][16 * LH];

  const int tid = threadIdx.x;
  const int wave = tid >> 5, lane = tid & 31;
  const int mbase = blockIdx.x * 16;

  for (int i = tid; i < 16 * HH; i += 128) {
    int r = i >> 7, c = i & (HH - 1);
    ldsH[r * LH + c] = h[(size_t)(mbase + r) * HH + c];
  }

  const int jt = blockIdx.y * 4 + wave;
  const bool active = jt < NTILES_J;
  const int jbase = jt * 16;
  if (active) {
    for (int i = lane; i < 16 * 32; i += 32) {
      int r = i >> 5, c4 = (i & 31) * 4;
      const float4 v = *(const float4*)&item_w[(size_t)(jbase + r) * HH + c4];
      *(float4*)&ldsB[wave][r * LH + c4] = v;
    }
  }
  __syncthreads();

  if (active) {
    const int ln = lane & 15, half = lane >> 4;
    v8f c = {};
#pragma unroll 8
    for (int k4 = 0; k4 < HH; k4 += 4) {
      v2f a = *(const v2f*)&ldsH[ln * LH + k4 + 2 * half];
      v2f b = *(const v2f*)&ldsB[wave][ln * LH + k4 + 2 * half];
      c = wmma_f32_k4(a, b, c);
    }
#pragma unroll
    for (int r = 0; r < 8; ++r)
      out[(size_t)(mbase + r + 8 * half) * II + jbase + ln] = c[r];
  }
}

// ---------------------------------------------------------------------------
// Host launcher
// ---------------------------------------------------------------------------
extern "C" void kernel_launch(void* const* d_in, const int* in_sizes, int n_in,
                              void* d_out, int out_size, void* d_ws, size_t ws_size,
                              hipStream_t stream) {
  (void)in_sizes; (void)n_in; (void)out_size; (void)ws_size;

  const int*   user          = (const int*)d_in[0];
  const int*   item          = (const int*)d_in[1];
  const int*   rating        = (const int*)d_in[2];
  const int*   item4user     = (const int*)d_in[3];
  const int*   social        = (const int*)d_in[4];
  const int*   social4user   = (const int*)d_in[5];
  const int*   bi_users      = (const int*)d_in[6];
  const int*   bi_ratings    = (const int*)d_in[7];
  const int*   bu_item       = (const int*)d_in[8];
  const int*   batch_target  = (const int*)d_in[9];
  const float* user_w   = (const float*)d_in[10];
  const float* item_w   = (const float*)d_in[11];
  const float* rating_w = (const float*)d_in[12];
  const float* gv_w  = (const float*)d_in[13]; const float* gv_b  = (const float*)d_in[14];
  const float* gu_w  = (const float*)d_in[15]; const float* gu_b  = (const float*)d_in[16];
  const float* eq5_w1  = (const float*)d_in[17]; const float* eq5_b1  = (const float*)d_in[18];
  const float* eq5_w2  = (const float*)d_in[19]; const float* eq5_b2  = (const float*)d_in[20];
  const float* eq5i_w1 = (const float*)d_in[21]; const float* eq5i_b1 = (const float*)d_in[22];
  const float* eq5i_w2 = (const float*)d_in[23]; const float* eq5i_b2 = (const float*)d_in[24];
  const float* eq4_w   = (const float*)d_in[25]; const float* eq4_b   = (const float*)d_in[26];
  const float* eq4i_w  = (const float*)d_in[27]; const float* eq4i_b  = (const float*)d_in[28];
  const float* eq10_w1 = (const float*)d_in[29]; const float* eq10_b1 = (const float*)d_in[30];
  const float* eq10_w2 = (const float*)d_in[31]; const float* eq10_b2 = (const float*)d_in[32];
  const float* eq9_w   = (const float*)d_in[33]; const float* eq9_b   = (const float*)d_in[34];
  const float* eq13_w  = (const float*)d_in[35]; const float* eq13_b  = (const float*)d_in[36];
  const float* mlp_w1  = (const float*)d_in[37]; const float* mlp_b1  = (const float*)d_in[38];
  const float* mlp_w2  = (const float*)d_in[39]; const float* mlp_b2  = (const float*)d_in[40];
  const float* mlp_w3  = (const float*)d_in[41]; const float* mlp_b3  = (const float*)d_in[42];

  char* wsp = (char*)d_ws;
  size_t off = 0;
  auto alloc = [&](size_t bytes) { size_t o = off; off = (off + bytes + 255) & ~(size_t)255; return o; };

  float* xia   = (float*)(wsp + alloc((size_t)EE1 * HH * 4));
  float* fjt   = (float*)(wsp + alloc((size_t)EE3 * HH * 4));
  float* sc1   = (float*)(wsp + alloc((size_t)EE1 * 4));
  float* sc2   = (float*)(wsp + alloc((size_t)EE3 * 4));
  float* sc3   = (float*)(wsp + alloc((size_t)EE2 * 4));
  int*   idx1  = (int*)  (wsp + alloc((size_t)EE1 * 4));   // user[item4user[e]]
  int*   idx2  = (int*)  (wsp + alloc((size_t)EE3 * 4));   // batch_target[bu_item[e]]
  int*   idx3  = (int*)  (wsp + alloc((size_t)EE2 * 4));   // user[social4user[e]]
  // contiguous zero-init region: segmax[3B] (uint) | segsum[3B] | acc[3][B*H]
  size_t zbytes = (size_t)(3 * BB + 3 * BB + 3 * BB * HH) * 4;
  char*  zbase  = wsp + alloc(zbytes);
  unsigned* segmax = (unsigned*)zbase;
  float*    segsum = (float*)(zbase + (size_t)3 * BB * 4);
  float*    acc    = (float*)(zbase + (size_t)6 * BB * 4);
  float* hi    = (float*)(wsp + alloc((size_t)BB * HH * 4));
  float* zj    = (float*)(wsp + alloc((size_t)BB * HH * 4));
  float* hs    = (float*)(wsp + alloc((size_t)BB * HH * 4));
  float* hcomb = (float*)(wsp + alloc((size_t)BB * HH * 4));
  float* s1    = (float*)(wsp + alloc((size_t)BB * HH * 4));
  float* s2    = (float*)(wsp + alloc((size_t)BB * HH * 4));

  float* out_score  = (float*)d_out;
  float* out_logits = out_score + BB;

  hipMemsetAsync(zbase, 0, zbytes, stream);

  // composed gather indices
  compose_idx_kernel<<<(EE1 + 255) / 256, 256, 0, stream>>>(item4user, user, idx1, EE1);
  compose_idx_kernel<<<(EE3 + 255) / 256, 256, 0, stream>>>(bu_item, batch_target, idx2, EE3);
  compose_idx_kernel<<<(EE2 + 255) / 256, 256, 0, stream>>>(social4user, user, idx3, EE2);

  // edge features
  gemm_gcat_kernel<256><<<EE1 / 16, 128, 0, stream>>>(item_w, item, rating_w, rating,
                                                      gv_w, gv_b, xia, 0);
  gemm_gcat_kernel<256><<<EE3 / 16, 128, 0, stream>>>(user_w, bi_users, rating_w, bi_ratings,
                                                      gu_w, gu_b, fjt, 0);

  // attention scores (fused 2-layer MLP)
  attn_score_kernel<<<EE1 / 16, 128, 0, stream>>>(xia, nullptr, user_w, idx1,
                                                  eq5_w1, eq5_b1, eq5_w2, eq5_b2, sc1);
  attn_score_kernel<<<EE3 / 16, 128, 0, stream>>>(fjt, nullptr, item_w, idx2,
                                                  eq5i_w1, eq5i_b1, eq5i_w2, eq5i_b2, sc2);
  attn_score_kernel<<<EE2 / 16, 128, 0, stream>>>(user_w, social, user_w, idx3,
                                                  eq10_w1, eq10_b1, eq10_w2, eq10_b2, sc3);

  // segment softmax
  seg_max_kernel<<<(EE1 + 255) / 256, 256, 0, stream>>>(sc1, item4user,   segmax + 0 * BB, EE1);
  seg_max_kernel<<<(EE3 + 255) / 256, 256, 0, stream>>>(sc2, bu_item,     segmax + 1 * BB, EE3);
  seg_max_kernel<<<(EE2 + 255) / 256, 256, 0, stream>>>(sc3, social4user, segmax + 2 * BB, EE2);
  seg_expsum_kernel<<<(EE1 + 255) / 256, 256, 0, stream>>>(sc1, item4user,   segmax + 0 * BB, segsum + 0 * BB, EE1);
  seg_expsum_kernel<<<(EE3 + 255) / 256, 256, 0, stream>>>(sc2, bu_item,     segmax + 1 * BB, segsum + 1 * BB, EE3);
  seg_expsum_kernel<<<(EE2 + 255) / 256, 256, 0, stream>>>(sc3, social4user, segmax + 2 * BB, segsum + 2 * BB, EE2);

  // weighted segment sums
  seg_scatter_kernel<<<EE1 * HH / 256, 256, 0, stream>>>(xia, nullptr, sc1, segsum + 0 * BB,
                                                         item4user, acc + (size_t)0 * BB * HH, EE1);
  seg_scatter_kernel<<<EE3 * HH / 256, 256, 0, stream>>>(fjt, nullptr, sc2, segsum + 1 * BB,
                                                         bu_item, acc + (size_t)1 * BB * HH, EE3);
  seg_scatter_kernel<<<EE2 * HH / 256, 256, 0, stream>>>(user_w, idx3, sc3, segsum + 2 * BB,
                                                         social4user, acc + (size_t)2 * BB * HH, EE2);

  // per-user heads
  gemm_gcat_kernel<128><<<BB / 16, 128, 0, stream>>>(acc + (size_t)0 * BB * HH, nullptr, nullptr, nullptr,
                                                     eq4_w, eq4_b, hi, 1);
  gemm_gcat_kernel<128><<<BB / 16, 128, 0, stream>>>(acc + (size_t)1 * BB * HH, nullptr, nullptr, nullptr,
                                                     eq4i_w, eq4i_b, zj, 1);
  gemm_gcat_kernel<128><<<BB / 16, 128, 0, stream>>>(acc + (size_t)2 * BB * HH, nullptr, nullptr, nullptr,
                                                     eq9_w, eq9_b, hs, 1);
  gemm_gcat_kernel<256><<<BB / 16, 128, 0, stream>>>(hi, nullptr, hs, nullptr,
                                                     eq13_w, eq13_b, hcomb, 1);
  gemm_gcat_kernel<256><<<BB / 16, 128, 0, stream>>>(hcomb, nullptr, zj, nullptr,
                                                     mlp_w1, mlp_b1, s1, 1);
  gemm_gcat_kernel<128><<<BB / 16, 128, 0, stream>>>(s1, nullptr, nullptr, nullptr,
                                                     mlp_w2, mlp_b2, s2, 1);
  rowdot_kernel<<<BB / 8, dim3(32, 8), 0, stream>>>(s2, mlp_w3, mlp_b3, out_score, BB);

  // big GEMM: logits = hcomb @ item_w^T
  dim3 lgrid(BB / 16, (NTILES_J + 3) / 4);
  logits_kernel<<<lgrid, 128, 0, stream>>>(hcomb, item_w, out_logits);
}